// VGAEModel_33809982554538
// MI455X (gfx1250) — compile-verified
//
#include <hip/hip_runtime.h>
#include <hip/hip_bf16.h>

typedef float v2f __attribute__((ext_vector_type(2)));
typedef float v8f __attribute__((ext_vector_type(8)));

// ---------------------------------------------------------------------------
// Degree count: deg[dst[e]] += 1   (native global_atomic_add_u32)
// ---------------------------------------------------------------------------
__global__ void count_deg_kernel(const int* __restrict__ dst,
                                 unsigned* __restrict__ deg, int E) {
    int i = blockIdx.x * blockDim.x + threadIdx.x;
    if (i < E) atomicAdd(&deg[dst[i]], 1u);
}

// dinv[n] = rsqrt(deg[n] + 1)   (+1 = self loop; always > 0)
__global__ void dinv_kernel(const unsigned* __restrict__ deg,
                            float* __restrict__ dinv, int N) {
    int i = blockIdx.x * blockDim.x + threadIdx.x;
    if (i < N) dinv[i] = rsqrtf((float)(deg[i] + 1u));
}

// ---------------------------------------------------------------------------
// Single-block exclusive scan of deg -> offs[0..N], cursor copy.
// 1024 threads; each thread serially reduces/scans a contiguous chunk,
// chunk sums scanned in LDS (Hillis-Steele).
// ---------------------------------------------------------------------------
__global__ __launch_bounds__(1024) void scan_kernel(const unsigned* __restrict__ deg,
                                                    unsigned* __restrict__ offs,
                                                    unsigned* __restrict__ cursor,
                                                    int N) {
    __shared__ unsigned sums[1024];
    const int t   = threadIdx.x;
    const int per = (N + 1023) / 1024;
    const int lo  = t * per;
    const int hi  = min(N, lo + per);

    unsigned s = 0;
    for (int i = lo; i < hi; ++i) s += deg[i];
    sums[t] = s;
    __syncthreads();

    for (int off = 1; off < 1024; off <<= 1) {
        unsigned v = (t >= off) ? sums[t - off] : 0u;
        __syncthreads();
        sums[t] += v;
        __syncthreads();
    }

    unsigned base = (t == 0) ? 0u : sums[t - 1];
    for (int i = lo; i < hi; ++i) {
        offs[i]   = base;
        cursor[i] = base;
        base += deg[i];
    }
    if (t == 1023) offs[N] = base;  // == E
}

// ---------------------------------------------------------------------------
// CSR fill: csr_src[pos++] = src for each in-edge of dst
// ---------------------------------------------------------------------------
__global__ void csr_fill_kernel(const int* __restrict__ src,
                                const int* __restrict__ dst,
                                unsigned* __restrict__ cursor,
                                int* __restrict__ csr_src, int E) {
    int e = blockIdx.x * blockDim.x + threadIdx.x;
    if (e < E) {
        unsigned pos = atomicAdd(&cursor[dst[e]], 1u);
        csr_src[pos] = src[e];
    }
}

// ---------------------------------------------------------------------------
// fp32 GEMM via V_WMMA_F32_16X16X4_F32 (verified codegen in round 1).
// One wave32 computes one 16x16 tile; block = 8 waves = 128(M) x 16(N).
// 16-column slice of W staged in LDS.
// ---------------------------------------------------------------------------
template <int KDIM, int NCOLS>
__launch_bounds__(256)
__global__ void gemm_wmma_f32(const float* __restrict__ X,
                              const float* __restrict__ W,
                              float* __restrict__ out, int nrows) {
    __shared__ float Ws[KDIM * 16];

    const int tid     = threadIdx.x;
    const int lane    = tid & 31;
    const int waveId  = tid >> 5;
    const int colBase = blockIdx.y * 16;

    for (int idx = tid; idx < KDIM * 16; idx += 256) {
        int k = idx >> 4, n = idx & 15;
        Ws[idx] = W[k * NCOLS + colBase + n];
    }
    __syncthreads();

    const int rowBase = (blockIdx.x * 8 + waveId) * 16;
    if (rowBase + 16 > nrows) return;  // wave-uniform: EXEC stays all-ones

    const int m    = lane & 15;
    const int n    = lane & 15;
    const int half = lane >> 4;

    const float* xrow = X + (size_t)(rowBase + m) * KDIM;

    v8f acc = {};
#pragma unroll 4
    for (int k0 = 0; k0 < KDIM; k0 += 4) {
        const int ka = k0 + 2 * half;
        v2f a = *(const v2f*)(xrow + ka);
        v2f b;
        b.x = Ws[ka * 16 + n];
        b.y = Ws[(ka + 1) * 16 + n];
        acc = __builtin_amdgcn_wmma_f32_16x16x4_f32(
            false, a, false, b, (short)0, acc, false, false);
    }

#pragma unroll
    for (int v = 0; v < 8; ++v) {
        const int row = rowBase + v + 8 * half;
        out[(size_t)row * NCOLS + colBase + n] = acc[v];
    }
}

// ---------------------------------------------------------------------------
// Gather propagate, 128-wide, fused self-loop + bias + ReLU.
// One wave32 per destination node; lane owns features [4L .. 4L+3].
// out[n] = relu( tmp[n]*dinv[n]^2 + sum_{s in in(n)} tmp[s]*dinv[s]*dinv[n] + b )
// ---------------------------------------------------------------------------
__global__ void gather_f128_kernel(const unsigned* __restrict__ offs,
                                   const int* __restrict__ csr_src,
                                   const float* __restrict__ dinv,
                                   const float* __restrict__ tmp,
                                   const float* __restrict__ bias,
                                   float* __restrict__ out, int N) {
    int gid  = blockIdx.x * blockDim.x + threadIdx.x;
    int node = gid >> 5;
    int lane = gid & 31;
    if (node >= N) return;

    const float    dn  = dinv[node];
    const unsigned beg = offs[node];
    const unsigned end = offs[node + 1];

    // self-loop contribution
    float4 self = ((const float4*)(tmp + (size_t)node * 128))[lane];
    float ax = self.x * dn * dn, ay = self.y * dn * dn,
          az = self.z * dn * dn, aw = self.w * dn * dn;

    unsigned i = beg;
    for (; i + 2 <= end; i += 2) {             // 2 row-reads in flight
        int s0 = csr_src[i], s1 = csr_src[i + 1];
        float n0 = dinv[s0] * dn, n1 = dinv[s1] * dn;
        float4 v0 = ((const float4*)(tmp + (size_t)s0 * 128))[lane];
        float4 v1 = ((const float4*)(tmp + (size_t)s1 * 128))[lane];
        ax += v0.x * n0 + v1.x * n1;
        ay += v0.y * n0 + v1.y * n1;
        az += v0.z * n0 + v1.z * n1;
        aw += v0.w * n0 + v1.w * n1;
    }
    if (i < end) {
        int s = csr_src[i];
        float nr = dinv[s] * dn;
        float4 v = ((const float4*)(tmp + (size_t)s * 128))[lane];
        ax += v.x * nr; ay += v.y * nr; az += v.z * nr; aw += v.w * nr;
    }

    float4 b = ((const float4*)bias)[lane];
    float4 r;
    r.x = fmaxf(ax + b.x, 0.0f);
    r.y = fmaxf(ay + b.y, 0.0f);
    r.z = fmaxf(az + b.z, 0.0f);
    r.w = fmaxf(aw + b.w, 0.0f);
    ((float4*)(out + (size_t)node * 128))[lane] = r;
}

// ---------------------------------------------------------------------------
// Fused gather for mu + logstd (64-wide each), self-loop + bias included.
// One wave32 per node; lane owns features [2L, 2L+1] of both outputs.
// Edge list / dinv read once for both heads.
// ---------------------------------------------------------------------------
__global__ void gather_mu_ls_kernel(const unsigned* __restrict__ offs,
                                    const int* __restrict__ csr_src,
                                    const float* __restrict__ dinv,
                                    const float* __restrict__ tmu,
                                    const float* __restrict__ tls,
                                    const float* __restrict__ bmu,
                                    const float* __restrict__ bls,
                                    float* __restrict__ omu,
                                    float* __restrict__ ols, int N) {
    int gid  = blockIdx.x * blockDim.x + threadIdx.x;
    int node = gid >> 5;
    int lane = gid & 31;
    if (node >= N) return;

    const float    dn  = dinv[node];
    const unsigned beg = offs[node];
    const unsigned end = offs[node + 1];

    float2 sm = ((const float2*)(tmu + (size_t)node * 64))[lane];
    float2 sl = ((const float2*)(tls + (size_t)node * 64))[lane];
    float mx = sm.x * dn * dn, my = sm.y * dn * dn;
    float lx = sl.x * dn * dn, ly = sl.y * dn * dn;

    unsigned i = beg;
    for (; i + 2 <= end; i += 2) {
        int s0 = csr_src[i], s1 = csr_src[i + 1];
        float n0 = dinv[s0] * dn, n1 = dinv[s1] * dn;
        float2 m0 = ((const float2*)(tmu + (size_t)s0 * 64))[lane];
        float2 l0 = ((const float2*)(tls + (size_t)s0 * 64))[lane];
        float2 m1 = ((const float2*)(tmu + (size_t)s1 * 64))[lane];
        float2 l1 = ((const float2*)(tls + (size_t)s1 * 64))[lane];
        mx += m0.x * n0 + m1.x * n1;
        my += m0.y * n0 + m1.y * n1;
        lx += l0.x * n0 + l1.x * n1;
        ly += l0.y * n0 + l1.y * n1;
    }
    if (i < end) {
        int s = csr_src[i];
        float nr = dinv[s] * dn;
        float2 m0 = ((const float2*)(tmu + (size_t)s * 64))[lane];
        float2 l0 = ((const float2*)(tls + (size_t)s * 64))[lane];
        mx += m0.x * nr; my += m0.y * nr;
        lx += l0.x * nr; ly += l0.y * nr;
    }

    float2 bm = ((const float2*)bmu)[lane];
    float2 bl = ((const float2*)bls)[lane];
    float2 rm; rm.x = mx + bm.x; rm.y = my + bm.y;
    float2 rl; rl.x = lx + bl.x; rl.y = ly + bl.y;
    ((float2*)(omu + (size_t)node * 64))[lane] = rm;
    ((float2*)(ols + (size_t)node * 64))[lane] = rl;
}

// ---------------------------------------------------------------------------
// Host-side launcher
// ---------------------------------------------------------------------------
extern "C" void kernel_launch(void* const* d_in, const int* in_sizes, int n_in,
                              void* d_out, int out_size, void* d_ws, size_t ws_size,
                              hipStream_t stream) {
    constexpr int IN = 256, HID = 128, OUT = 64;

    const float* x    = (const float*)d_in[0];
    const int*   eidx = (const int*)d_in[1];
    const float* W1   = (const float*)d_in[3];
    const float* b1   = (const float*)d_in[4];
    const float* Wmu  = (const float*)d_in[5];
    const float* bmu  = (const float*)d_in[6];
    const float* Wls  = (const float*)d_in[7];
    const float* bls  = (const float*)d_in[8];

    const int N = in_sizes[0] / IN;   // 50000
    const int E = in_sizes[1] / 2;    // 1600000
    const int* src = eidx;
    const int* dst = eidx + E;

    // Workspace carve-up
    char* ws = (char*)d_ws;
    unsigned* deg    = (unsigned*)ws;  ws += (size_t)N * sizeof(unsigned);
    float*    dinv   = (float*)ws;     ws += (size_t)N * sizeof(float);
    unsigned* offs   = (unsigned*)ws;  ws += (size_t)(N + 1) * sizeof(unsigned);
    unsigned* cursor = (unsigned*)ws;  ws += (size_t)N * sizeof(unsigned);
    int*      csrS   = (int*)ws;       ws += (size_t)E * sizeof(int);
    float*    tmp1   = (float*)ws;     ws += (size_t)N * HID * sizeof(float);
    float*    h      = (float*)ws;     ws += (size_t)N * HID * sizeof(float);
    float*    tmu    = (float*)ws;     ws += (size_t)N * OUT * sizeof(float);
    float*    tls    = (float*)ws;     /* end */

    float* omu = (float*)d_out;
    float* ols = (float*)d_out + (size_t)N * OUT;

    const int TB = 256;

    // 1) degrees, norm factors, CSR build (gather-form graph)
    hipMemsetAsync(deg, 0, (size_t)N * sizeof(unsigned), stream);
    count_deg_kernel<<<(E + TB - 1) / TB, TB, 0, stream>>>(dst, deg, E);
    dinv_kernel<<<(N + TB - 1) / TB, TB, 0, stream>>>(deg, dinv, N);
    scan_kernel<<<1, 1024, 0, stream>>>(deg, offs, cursor, N);
    csr_fill_kernel<<<(E + TB - 1) / TB, TB, 0, stream>>>(src, dst, cursor, csrS, E);

    // 2) layer 1: tmp1 = x @ W1  (WMMA f32), then gather+bias+relu -> h
    {
        int rowTiles = (N + 15) / 16;
        dim3 grid((rowTiles + 7) / 8, HID / 16);
        gemm_wmma_f32<IN, HID><<<grid, TB, 0, stream>>>(x, W1, tmp1, N);
    }
    gather_f128_kernel<<<((size_t)N * 32 + TB - 1) / TB, TB, 0, stream>>>(
        offs, csrS, dinv, tmp1, b1, h, N);

    // 3) mu / logstd GEMMs (WMMA f32)
    {
        int rowTiles = (N + 15) / 16;
        dim3 grid((rowTiles + 7) / 8, OUT / 16);
        gemm_wmma_f32<HID, OUT><<<grid, TB, 0, stream>>>(h, Wmu, tmu, N);
        gemm_wmma_f32<HID, OUT><<<grid, TB, 0, stream>>>(h, Wls, tls, N);
    }

    // 4) fused gather for both heads, bias included, straight into d_out
    gather_mu_ls_kernel<<<((size_t)N * 32 + TB - 1) / TB, TB, 0, stream>>>(
        offs, csrS, dinv, tmu, tls, bmu, bls, omu, ols, N);
}